// NonIsotropicNC_73469710566014
// MI455X (gfx1250) — compile-verified
//
#include <hip/hip_runtime.h>
#include <hip/hip_bf16.h>

// ---------------------------------------------------------------------------
// NonIsotropicNC 4D-conv net on MI455X (gfx1250), fp32 via V_WMMA_F32_16X16X4_F32.
//
// Math reduction (exact):
//   _sym(x,w,b) = relu(conv4d(x,w)+b) + P(relu(conv4d(Px,w)+b))
//               = relu(conv4d(x,wA)+b) + relu(conv4d(x,wB)+b)
// where wA = w zero-padded to (5,5,5,5) on the first spatial pair and
// wB[o,c,i,j,p,q] = wA[o,c,p,q,i,j]. Each layer's concat folds into O=16.
// ---------------------------------------------------------------------------

typedef float v2f __attribute__((ext_vector_type(2)));
typedef float v8f __attribute__((ext_vector_type(8)));

#define KPAD 628            // 625 taps per channel, padded to multiple of 4
#define KSTEPS 157          // KPAD/4

// ---------------------------------------------------------------------------
// Weight prep: build WA/WB in [c][kk][o] layout (o fastest, 16-wide) so the
// WMMA A-fragment per lane is two contiguous-row dword loads.
// ---------------------------------------------------------------------------
__global__ __launch_bounds__(256)
void prep_weights(const float* __restrict__ src0, const float* __restrict__ b0,
                  const float* __restrict__ src1, const float* __restrict__ b1,
                  float* __restrict__ WA, float* __restrict__ WB,
                  float* __restrict__ biasOut,
                  int C, int ka0, int ka1, int O0)
{
    int t = blockIdx.x * blockDim.x + threadIdx.x;
    if (t < 16) {
        float bz = 0.f;
        if (t < 8) { if (t < O0) bz = b0[t]; }
        else       { if (src1)   bz = b1[t - 8]; }
        biasOut[t] = bz;
    }
    int total = C * KPAD * 16;
    if (t >= total) return;

    int o  = t & 15;
    int kk = (t >> 4) % KPAD;
    int c  = (t >> 4) / KPAD;

    float vA = 0.f, vB = 0.f;
    if (kk < 625) {
        int ij = kk / 25, pq = kk - ij * 25;
        int i = ij / 5, j = ij - i * 5;
        int p = pq / 5, q = pq - p * 5;

        const float* src = nullptr; int ka = 5; int oo = 0;
        if (o < 8) { if (o < O0)  { src = src0; ka = ka0; oo = o; } }
        else       { if (src1)    { src = src1; ka = ka1; oo = o - 8; } }

        if (src) {
            int off = (5 - ka) >> 1;
            int ia = i - off, ja = j - off;
            if (ia >= 0 && ia < ka && ja >= 0 && ja < ka)
                vA = src[((((oo * C + c) * ka + ia) * ka + ja) * 5 + p) * 5 + q];
            int pb = p - off, qb = q - off;
            if (pb >= 0 && pb < ka && qb >= 0 && qb < ka)
                vB = src[((((oo * C + c) * ka + pb) * ka + qb) * 5 + i) * 5 + j];
        }
    }
    WA[t] = vA;   // t == (c*KPAD + kk)*16 + o
    WB[t] = vB;
}

// ---------------------------------------------------------------------------
// Sym-conv layer: out = relu(convA(in)+b) + relu(convB(in)+b)
// One workgroup per (b,h1,w1) tile; 256 threads = 8 waves; wave owns
// N-subtiles s = 2*wave+{0,1} (16x16 O x w2 tiles, h2 = s).
// LDS: haloed input window (25 x 20x20, zero border) filled via
// global_load_async_to_lds_b32 (ASYNCcnt path), plus a precomputed
// k -> LDS-base address table to strip VALU from the k-loop.
// ---------------------------------------------------------------------------
__global__ __launch_bounds__(256)
void conv4d_sym(const float* __restrict__ in,
                const float* __restrict__ WA,
                const float* __restrict__ WB,
                const float* __restrict__ bias,
                float* __restrict__ out,
                int C, int Ovalid)
{
    __shared__ float lds[10000];          // 25 * 20 * 20 floats (~39 KB)
    __shared__ int   adTab[KPAD];         // k -> ij*400 + p*20 + q

    const int tid  = threadIdx.x;
    const int wave = tid >> 5;
    const int lane = tid & 31;
    const int lh   = lane >> 4;           // lane-half (K offset +2)
    const int col  = lane & 15;           // M (A) / N (B,C,D) index

    const int wg = blockIdx.x;            // 1024 = B(4) * H1(16) * W1(16)
    const int bb = wg >> 8;
    const int h1 = (wg >> 4) & 15;
    const int w1 = wg & 15;

    // ---- one-time init: zero window (border stays 0 forever; the valid
    // element set depends only on (h1,w1), identical across channels) and
    // build the k -> address-base table.
    for (int e = tid; e < 10000; e += 256) lds[e] = 0.f;
    for (int k = tid; k < KPAD; k += 256) {
        int kc = k > 624 ? 624 : k;       // pad taps alias k=624 (weight==0)
        int ij = kc / 25, pq = kc - ij * 25;
        int p  = pq / 5,  q  = pq - p * 5;
        adTab[k] = ij * 400 + p * 20 + q;
    }

    v8f accA[2] = {};
    v8f accB[2] = {};

    for (int c = 0; c < C; ++c) {
        __syncthreads();                  // all reads of previous slice done
        // ------ async-stage input window: copy this channel's haloed input from global into lds as [ij][y][x]
        const int chanBase = (bb * C + c) * 65536;
        for (int e = tid; e < 10000; e += 256) {
            int ij  = e / 400; int rem = e - ij * 400;
            int y   = rem / 20; int x = rem - y * 20;
            int i   = ij / 5;   int j = ij - i * 5;
            int h1s = h1 + i - 2, w1s = w1 + j - 2;
            int h2s = y - 2,      w2s = x - 2;
            if (((unsigned)h1s < 16u) & ((unsigned)w1s < 16u) &
                ((unsigned)h2s < 16u) & ((unsigned)w2s < 16u)) {
                unsigned ldsAddr =
                    (unsigned)(unsigned long long)(const void*)&lds[e];
                int goff = (chanBase + ((h1s * 16 + w1s) << 8)
                            + (h2s << 4) + w2s) * 4;
                asm volatile(
                    "global_load_async_to_lds_b32 %0, %1, %2 offset:0"
                    :: "v"(ldsAddr), "v"(goff), "s"(in) : "memory");
            }
        }
        asm volatile("s_wait_asynccnt 0x0" ::: "memory");
        __syncthreads();

        const float* wArow = WA + (c * KPAD) * 16;
        const float* wBrow = WB + (c * KPAD) * 16;

        // unroll 1: keeps the four 8-VGPR accumulators in fixed even-aligned
        // ranges; unrolling makes the RA ping-pong them with v_mov_b64 churn.
#pragma unroll 1
        for (int kstep = 0; kstep < KSTEPS; ++kstep) {
            const int kl = (kstep << 2) + (lh << 1);
            // A-fragments (weights), [K][16] layout: lane m=col, rows kl,kl+1
            v2f aA, aB;
            aA.x = wArow[(kl    ) * 16 + col];
            aA.y = wArow[(kl + 1) * 16 + col];
            aB.x = wBrow[(kl    ) * 16 + col];
            aB.y = wBrow[(kl + 1) * 16 + col];

            // B-fragment LDS bases from table (kl even -> 8B aligned pair)
            const int2 tt = *(const int2*)(adTab + kl);
            const int ad0 = tt.x + col;
            const int ad1 = tt.y + col;

#pragma unroll
            for (int si = 0; si < 2; ++si) {
                const int s = (wave << 1) + si;             // h2 row
                v2f bf;
                bf.x = lds[ad0 + s * 20];
                bf.y = lds[ad1 + s * 20];
                accA[si] = __builtin_amdgcn_wmma_f32_16x16x4_f32(
                    false, aA, false, bf, (short)0, accA[si], false, false);
                accB[si] = __builtin_amdgcn_wmma_f32_16x16x4_f32(
                    false, aB, false, bf, (short)0, accB[si], false, false);
            }
        }
    }

    // ------ epilogue: bias, dual ReLU, sum, store
#pragma unroll
    for (int si = 0; si < 2; ++si) {
        const int s = (wave << 1) + si;
#pragma unroll
        for (int r = 0; r < 8; ++r) {
            const int m = r + (lh << 3);                    // output channel
            const float bz = bias[m];
            float vA = accA[si][r] + bz; vA = vA > 0.f ? vA : 0.f;
            float vB = accB[si][r] + bz; vB = vB > 0.f ? vB : 0.f;
            if (m < Ovalid)
                out[(bb * Ovalid + m) * 65536 + ((h1 * 16 + w1) << 8)
                    + (s << 4) + col] = vA + vB;
        }
    }
}

// ---------------------------------------------------------------------------
extern "C" void kernel_launch(void* const* d_in, const int* in_sizes, int n_in,
                              void* d_out, int out_size, void* d_ws, size_t ws_size,
                              hipStream_t stream) {
    (void)in_sizes; (void)n_in; (void)out_size; (void)ws_size;
    const float* x   = (const float*)d_in[0];
    const float* w00 = (const float*)d_in[1];
    const float* b00 = (const float*)d_in[2];
    const float* w01 = (const float*)d_in[3];
    const float* b01 = (const float*)d_in[4];
    const float* w10 = (const float*)d_in[5];
    const float* b10 = (const float*)d_in[6];
    const float* w11 = (const float*)d_in[7];
    const float* b11 = (const float*)d_in[8];
    const float* w2  = (const float*)d_in[9];
    const float* b2  = (const float*)d_in[10];

    float* ws = (float*)d_ws;
    float* mid1  = ws;                       // 4*16*65536
    float* mid2  = mid1 + 4194304;           // 4*16*65536
    float* W1A   = mid2 + 4194304;           // 1*628*16
    float* W1B   = W1A + KPAD * 16;
    float* W2A   = W1B + KPAD * 16;          // 16*628*16
    float* W2B   = W2A + 16 * KPAD * 16;
    float* W3A   = W2B + 16 * KPAD * 16;
    float* W3B   = W3A + 16 * KPAD * 16;
    float* bias1 = W3B + 16 * KPAD * 16;
    float* bias2 = bias1 + 16;
    float* bias3 = bias2 + 16;
    // total ws usage: ~36.2 MB

    const dim3 blk(256);
    prep_weights<<<dim3((1 * KPAD * 16 + 255) / 256), blk, 0, stream>>>(
        w00, b00, w01, b01, W1A, W1B, bias1, /*C=*/1, /*ka0=*/3, /*ka1=*/5, /*O0=*/8);
    prep_weights<<<dim3((16 * KPAD * 16 + 255) / 256), blk, 0, stream>>>(
        w10, b10, w11, b11, W2A, W2B, bias2, /*C=*/16, 3, 5, 8);
    prep_weights<<<dim3((16 * KPAD * 16 + 255) / 256), blk, 0, stream>>>(
        w2, b2, nullptr, nullptr, W3A, W3B, bias3, /*C=*/16, 5, 5, /*O0=*/1);

    const dim3 grid(1024);   // B * H1 * W1
    conv4d_sym<<<grid, blk, 0, stream>>>(x,    W1A, W1B, bias1, mid1,          1, 16);
    conv4d_sym<<<grid, blk, 0, stream>>>(mid1, W2A, W2B, bias2, mid2,         16, 16);
    conv4d_sym<<<grid, blk, 0, stream>>>(mid2, W3A, W3B, bias3, (float*)d_out, 16, 1);
}